// GNN_27745488732760
// MI455X (gfx1250) — compile-verified
//
#include <hip/hip_runtime.h>

typedef float v2f __attribute__((ext_vector_type(2)));
typedef float v8f __attribute__((ext_vector_type(8)));

#define NIN 128
#define NOUT 64

// ---------------------------------------------------------------------------
// Degree / normalization
// ---------------------------------------------------------------------------
__global__ void k_init_deg(float* deg, int n) {
    int i = blockIdx.x * blockDim.x + threadIdx.x;
    if (i < n) deg[i] = 1.0f;  // self-loop contributes 1
}

__global__ void k_deg_accum(const int* __restrict__ dst, float* deg, int e) {
    int i = blockIdx.x * blockDim.x + threadIdx.x;
    if (i < e) atomicAdd(&deg[dst[i]], 1.0f);
}

__global__ void k_deg_to_dinv(float* deg, int n) {
    int i = blockIdx.x * blockDim.x + threadIdx.x;
    if (i < n) deg[i] = rsqrtf(deg[i]);  // deg >= 1 always (self-loop)
}

// ---------------------------------------------------------------------------
// fp32 GEMM via V_WMMA_F32_16X16X4_F32.
// A: [N,K] row-major, B: [K,F] row-major, C: [N,F] = A@B (+bias).
// One wave computes a 16-row strip across ALL F columns (F/16 accumulators),
// so A is read once per row tile and B (tiny) stays hot in L2/L0.
// A 16x4 layout: lanes 0-15 hold M=lane, K={k0,k0+1}; lanes 16-31 K={k0+2,k0+3}.
// B 4x16 layout: VGPR j holds row K=k0+2*(lane/16)+j, col N=lane%16.
// C 16x16 layout: VGPR r holds M=r+8*(lane/16), N=lane%16.
// ---------------------------------------------------------------------------
template <int K, int F>
__global__ void k_gemm_wmma(const float* __restrict__ A,
                            const float* __restrict__ B,
                            const float* __restrict__ bias,
                            float* __restrict__ C, int nrows) {
    constexpr int FT = F / 16;
    const int wave = threadIdx.x >> 5;
    const int lane = threadIdx.x & 31;
    const int rowTile = blockIdx.x * (blockDim.x >> 5) + wave;
    if (rowTile * 16 >= nrows) return;  // wave-uniform: EXEC stays all-ones

    const int half = lane >> 4;
    const int l16  = lane & 15;

    v8f zero = {};
    v8f acc[FT];
#pragma unroll
    for (int ft = 0; ft < FT; ++ft) acc[ft] = zero;

    const float* arow = A + (long)(rowTile * 16 + l16) * K + 2 * half;

    for (int k0 = 0; k0 < K; k0 += 4) {
        float2 av = *(const float2*)(arow + k0);
        v2f a; a.x = av.x; a.y = av.y;
        const int kb = k0 + 2 * half;
#pragma unroll
        for (int ft = 0; ft < FT; ++ft) {
            v2f b;
            b.x = B[(long)kb * F + ft * 16 + l16];
            b.y = B[(long)(kb + 1) * F + ft * 16 + l16];
            acc[ft] = __builtin_amdgcn_wmma_f32_16x16x4_f32(
                false, a, false, b, (short)0, acc[ft], false, false);
        }
    }

#pragma unroll
    for (int ft = 0; ft < FT; ++ft) {
        const float bv = bias ? bias[ft * 16 + l16] : 0.0f;
#pragma unroll
        for (int r = 0; r < 8; ++r) {
            const int row = rowTile * 16 + r + 8 * half;
            C[(long)row * F + ft * 16 + l16] = acc[ft][r] + bv;
        }
    }
}

// ---------------------------------------------------------------------------
// agg[i,f] = h[i,f] * dinv[i]^2 + b_gcn[f]   (self-loop term + post-agg bias)
// ---------------------------------------------------------------------------
__global__ void k_agg_init(const float* __restrict__ h,
                           const float* __restrict__ dinv,
                           const float* __restrict__ bias,
                           float* __restrict__ agg, long total) {
    long i = (long)blockIdx.x * blockDim.x + threadIdx.x;
    if (i >= total) return;
    int node = (int)(i >> 7);          // /128
    int f    = (int)(i & 127);
    float d = dinv[node];
    agg[i] = h[i] * d * d + bias[f];
}

// ---------------------------------------------------------------------------
// Edge scatter: one wave (32 lanes) per edge, 4 features per lane (128 total).
// float4 gather from h[src], 4 global f32 atomic adds into agg[dst].
// ---------------------------------------------------------------------------
__global__ void k_edge_scatter(const float* __restrict__ h,
                               const int* __restrict__ src,
                               const int* __restrict__ dst,
                               const float* __restrict__ dinv,
                               float* __restrict__ agg, int e) {
    long t = (long)blockIdx.x * blockDim.x + threadIdx.x;
    int edge = (int)(t >> 5);
    if (edge >= e) return;
    int lane = (int)(t & 31);
    int s = src[edge];
    int d = dst[edge];
    float nrm = dinv[s] * dinv[d];
    float4 v = *(const float4*)(h + (long)s * NIN + lane * 4);
    float* base = agg + (long)d * NIN + lane * 4;
    atomicAdd(base + 0, v.x * nrm);
    atomicAdd(base + 1, v.y * nrm);
    atomicAdd(base + 2, v.z * nrm);
    atomicAdd(base + 3, v.w * nrm);
}

// ---------------------------------------------------------------------------
// BatchNorm statistics: per-column sum & sum-of-squares.
// blockDim.x == F; each thread owns one column, strides over rows (coalesced).
// ---------------------------------------------------------------------------
__global__ void k_zero(float* p, int n) {
    int i = blockIdx.x * blockDim.x + threadIdx.x;
    if (i < n) p[i] = 0.0f;
}

__global__ void k_col_stats(const float* __restrict__ X, int nrows, int F,
                            float* __restrict__ stats) {
    int c = threadIdx.x;
    float s = 0.0f, s2 = 0.0f;
    for (int r = blockIdx.x; r < nrows; r += gridDim.x) {
        float v = X[(long)r * F + c];
        s += v;
        s2 += v * v;
    }
    atomicAdd(&stats[c], s);
    atomicAdd(&stats[F + c], s2);
}

// ---------------------------------------------------------------------------
// Fused BatchNorm + ReLU, in place. var is biased (sumsq/N - mean^2), eps=1e-5.
// ---------------------------------------------------------------------------
__global__ void k_bn_relu(float* __restrict__ X, int nrows, int F,
                          const float* __restrict__ stats,
                          const float* __restrict__ gamma,
                          const float* __restrict__ beta, long total) {
    long i = (long)blockIdx.x * blockDim.x + threadIdx.x;
    if (i >= total) return;
    int c = (int)(i % F);
    float invn  = 1.0f / (float)nrows;
    float mean  = stats[c] * invn;
    float var   = stats[F + c] * invn - mean * mean;
    float scale = gamma[c] * rsqrtf(var + 1e-5f);
    float v = (X[i] - mean) * scale + beta[c];
    X[i] = v > 0.0f ? v : 0.0f;
}

// ---------------------------------------------------------------------------
// Launch
// ---------------------------------------------------------------------------
extern "C" void kernel_launch(void* const* d_in, const int* in_sizes, int n_in,
                              void* d_out, int out_size, void* d_ws, size_t ws_size,
                              hipStream_t stream) {
    const float* x        = (const float*)d_in[0];
    const int*   edges    = (const int*)d_in[1];
    const float* W_gcn    = (const float*)d_in[2];
    const float* b_gcn    = (const float*)d_in[3];
    const float* bn_gamma = (const float*)d_in[4];
    const float* bn_beta  = (const float*)d_in[5];
    const float* W1       = (const float*)d_in[6];
    const float* b1       = (const float*)d_in[7];
    const float* mlp_g    = (const float*)d_in[8];
    const float* mlp_b    = (const float*)d_in[9];
    const float* W2       = (const float*)d_in[10];
    const float* b2       = (const float*)d_in[11];

    const int N = in_sizes[0] / NIN;      // 100000
    const int E = in_sizes[1] / 2;        // 640000
    const int* src = edges;               // edge_index[0,:]
    const int* dst = edges + E;           // edge_index[1,:]

    // Workspace layout (bytes):
    //   h    : N*128*4   (also reused for h2 = N*64*4 after GCN block)
    //   agg  : N*128*4
    //   dinv : N*4
    //   stats: 384*4  (stats1 = 2*128, stats2 at offset 256 = 2*64)
    char* ws = (char*)d_ws;
    float* h     = (float*)(ws);
    float* agg   = (float*)(ws + (size_t)N * NIN * 4);
    float* dinv  = (float*)(ws + (size_t)N * NIN * 8);
    float* stats = (float*)(ws + (size_t)N * NIN * 8 + (size_t)N * 4);
    float* stats1 = stats;
    float* stats2 = stats + 2 * NIN;
    float* h2  = h;                        // reuse: h dead after edge scatter
    float* out = (float*)d_out;

    const int B256 = 256;
    const int rowTiles = (N + 15) / 16;
    const int gemmBlocks = (rowTiles + 7) / 8;   // 8 waves/block, 1 row-tile/wave

    // 1. degree -> dinv
    k_init_deg<<<(N + B256 - 1) / B256, B256, 0, stream>>>(dinv, N);
    k_deg_accum<<<(E + B256 - 1) / B256, B256, 0, stream>>>(dst, dinv, E);
    k_deg_to_dinv<<<(N + B256 - 1) / B256, B256, 0, stream>>>(dinv, N);

    // 2. h = x @ W_gcn   (WMMA fp32)
    k_gemm_wmma<NIN, NIN><<<gemmBlocks, B256, 0, stream>>>(x, W_gcn, nullptr, h, N);

    // 3. aggregate: self-loop init + edge scatter-add
    long totNF = (long)N * NIN;
    k_agg_init<<<(int)((totNF + B256 - 1) / B256), B256, 0, stream>>>(h, dinv, b_gcn, agg, totNF);
    long scatterThreads = (long)E * 32;
    k_edge_scatter<<<(int)((scatterThreads + B256 - 1) / B256), B256, 0, stream>>>(
        h, src, dst, dinv, agg, E);

    // 4. BN + ReLU on agg (128 cols)
    k_zero<<<2, 192, 0, stream>>>(stats, 384);
    k_col_stats<<<512, NIN, 0, stream>>>(agg, N, NIN, stats1);
    k_bn_relu<<<(int)((totNF + B256 - 1) / B256), B256, 0, stream>>>(
        agg, N, NIN, stats1, bn_gamma, bn_beta, totNF);

    // 5. h2 = agg @ W1 + b1   (WMMA fp32)
    k_gemm_wmma<NIN, NOUT><<<gemmBlocks, B256, 0, stream>>>(agg, W1, b1, h2, N);

    // 6. BN + ReLU on h2 (64 cols)
    long totNO = (long)N * NOUT;
    k_col_stats<<<512, NOUT, 0, stream>>>(h2, N, NOUT, stats2);
    k_bn_relu<<<(int)((totNO + B256 - 1) / B256), B256, 0, stream>>>(
        h2, N, NOUT, stats2, mlp_g, mlp_b, totNO);

    // 7. out = h2 @ W2 + b2   (WMMA fp32)
    k_gemm_wmma<NOUT, NOUT><<<gemmBlocks, B256, 0, stream>>>(h2, W2, b2, out, N);
}